// GCN_7825430413942
// MI455X (gfx1250) — compile-verified
//
#include <hip/hip_runtime.h>

#define HIDDEN 128

typedef __attribute__((ext_vector_type(2))) float v2f;
typedef __attribute__((ext_vector_type(4))) float v4f;
typedef __attribute__((ext_vector_type(8))) float v8f;

// ---------------- degree / symmetric norm ----------------
__global__ void zero_deg_kernel(int* __restrict__ deg, int n) {
  int i = blockIdx.x * blockDim.x + threadIdx.x;
  if (i < n) deg[i] = 0;
}

__global__ void degree_kernel(const int* __restrict__ dst, int E, int* __restrict__ deg) {
  int e = blockIdx.x * blockDim.x + threadIdx.x;
  if (e < E) atomicAdd(&deg[dst[e]], 1);
}

__global__ void dinv_kernel(const int* __restrict__ deg, float* __restrict__ dinv, int n) {
  int i = blockIdx.x * blockDim.x + threadIdx.x;
  if (i < n) dinv[i] = rsqrtf((float)(deg[i] + 1));  // +1 for self-loop
}

// ---------------- dense GEMM: Out[N,128] = A[N,128] @ W[128,128] (fp32 WMMA) ----
__global__ __launch_bounds__(256)
void gemm_wmma_kernel(const float* __restrict__ A, const float* __restrict__ W,
                      float* __restrict__ Out, int nrow) {
  __shared__ float Ws[HIDDEN * HIDDEN];  // 64 KB
  for (int i = threadIdx.x; i < HIDDEN * HIDDEN / 4; i += blockDim.x)
    ((v4f*)Ws)[i] = ((const v4f*)W)[i];
  __syncthreads();

  const int wave  = threadIdx.x >> 5;
  const int lane  = threadIdx.x & 31;
  const int lhalf = lane >> 4;   // 0: lanes 0-15, 1: lanes 16-31
  const int l16   = lane & 15;
  const int mtile = blockIdx.x * 8 + wave;     // one 16-row tile per wave
  const int row0  = mtile * 16;
  if (row0 + 16 > nrow) return;                // N is a multiple of 16

  v8f acc[8] = {};                             // 16x128 output per wave

  const float* aptr = A + (size_t)(row0 + l16) * HIDDEN;

  for (int kb = 0; kb < HIDDEN; kb += 4) {
    const int ka = kb + 2 * lhalf;             // A 16x4: half-wave splits K
    v2f a;
    a.x = aptr[ka];
    a.y = aptr[ka + 1];
#pragma unroll
    for (int nt = 0; nt < 8; ++nt) {
      const int n = nt * 16 + l16;             // B 4x16: N = lane%16
      v2f b;
      b.x = Ws[ka * HIDDEN + n];
      b.y = Ws[(ka + 1) * HIDDEN + n];
      acc[nt] = __builtin_amdgcn_wmma_f32_16x16x4_f32(
          /*neg_a=*/false, a, /*neg_b=*/false, b,
          /*c_mod=*/(short)0, acc[nt], /*reuse_a=*/false, /*reuse_b=*/false);
    }
  }

  // C/D layout: vgpr j -> M = j + 8*lhalf, N = nt*16 + l16
#pragma unroll
  for (int nt = 0; nt < 8; ++nt) {
#pragma unroll
    for (int j = 0; j < 8; ++j) {
      const int r = row0 + j + 8 * lhalf;
      Out[(size_t)r * HIDDEN + nt * 16 + l16] = acc[nt][j];
    }
  }
}

// ---------------- aggregation init: agg = lin * dinv^2 (self-loop term) ----
__global__ void initagg_kernel(const float* __restrict__ lin, const float* __restrict__ dinv,
                               float* __restrict__ agg, int total) {
  int idx = blockIdx.x * blockDim.x + threadIdx.x;
  if (idx < total) {
    float w = dinv[idx >> 7];
    agg[idx] = lin[idx] * w * w;
  }
}

// ---------------- edge scatter: agg[dst] += lin[src] * dinv[src]*dinv[dst] ----
__global__ __launch_bounds__(256)
void edge_scatter_kernel(const int* __restrict__ src, const int* __restrict__ dst,
                         const float* __restrict__ dinv, const float* __restrict__ lin,
                         float* __restrict__ agg, int E) {
  const int e = blockIdx.x * 8 + (threadIdx.x >> 5);  // one wave per edge
  if (e >= E) return;
  const int lane = threadIdx.x & 31;
  const int s = src[e];
  const int d = dst[e];
  const float w = dinv[s] * dinv[d];
  const float4 v = ((const float4*)(lin + (size_t)s * HIDDEN))[lane];
  float* op = agg + (size_t)d * HIDDEN + lane * 4;
  atomicAdd(op + 0, v.x * w);  // resolves in L2 (51.2MB buffer << 192MB L2)
  atomicAdd(op + 1, v.y * w);
  atomicAdd(op + 2, v.z * w);
  atomicAdd(op + 3, v.w * w);
}

// ---------------- bias + ReLU (in place) ----------------
__global__ void finalize_kernel(float* __restrict__ agg, const float* __restrict__ bias,
                                int total) {
  int idx = blockIdx.x * blockDim.x + threadIdx.x;
  if (idx < total) {
    float v = agg[idx] + bias[idx & (HIDDEN - 1)];
    agg[idx] = v > 0.f ? v : 0.f;
  }
}

// ---------------- head: gather batch rows, 128x2 linear, relu, log_softmax ----
__global__ __launch_bounds__(256)
void head_kernel(const float* __restrict__ feat, const int* __restrict__ bidx, int B,
                 const float* __restrict__ Wl, const float* __restrict__ bl,
                 float* __restrict__ out) {
  const int i = blockIdx.x * 8 + (threadIdx.x >> 5);  // one wave per batch row
  if (i >= B) return;
  const int lane = threadIdx.x & 31;
  const int node = bidx[i];
  const float4 h = ((const float4*)(feat + (size_t)node * HIDDEN))[lane];

  // feats_out (third output, offset 4*B)
  ((float4*)(out + 4 * (size_t)B + (size_t)i * HIDDEN))[lane] = h;

  const int k = lane * 4;
  float p0 = h.x * Wl[(k + 0) * 2 + 0] + h.y * Wl[(k + 1) * 2 + 0] +
             h.z * Wl[(k + 2) * 2 + 0] + h.w * Wl[(k + 3) * 2 + 0];
  float p1 = h.x * Wl[(k + 0) * 2 + 1] + h.y * Wl[(k + 1) * 2 + 1] +
             h.z * Wl[(k + 2) * 2 + 1] + h.w * Wl[(k + 3) * 2 + 1];
#pragma unroll
  for (int m = 16; m >= 1; m >>= 1) {       // wave32 reduction
    p0 += __shfl_xor(p0, m, 32);
    p1 += __shfl_xor(p1, m, 32);
  }
  if (lane == 0) {
    float z0 = p0 + bl[0]; z0 = z0 > 0.f ? z0 : 0.f;
    float z1 = p1 + bl[1]; z1 = z1 > 0.f ? z1 : 0.f;
    const float mx = fmaxf(z0, z1);
    const float lse = mx + logf(expf(z0 - mx) + expf(z1 - mx));
    out[(size_t)i * 2 + 0] = z0 - lse;            // logp
    out[(size_t)i * 2 + 1] = z1 - lse;
    out[2 * (size_t)B + (size_t)i * 2 + 0] = z0;  // raw "out"
    out[2 * (size_t)B + (size_t)i * 2 + 1] = z1;
  }
}

extern "C" void kernel_launch(void* const* d_in, const int* in_sizes, int n_in,
                              void* d_out, int out_size, void* d_ws, size_t ws_size,
                              hipStream_t stream) {
  const float* x    = (const float*)d_in[0];
  const int*   ei   = (const int*)d_in[1];
  const int*   bidx = (const int*)d_in[2];
  const float* W1   = (const float*)d_in[3];
  const float* b1   = (const float*)d_in[4];
  const float* W2   = (const float*)d_in[5];
  const float* b2   = (const float*)d_in[6];
  const float* Wl   = (const float*)d_in[7];
  const float* bl   = (const float*)d_in[8];
  float* out = (float*)d_out;

  const int N = in_sizes[0] / HIDDEN;
  const int E = in_sizes[1] / 2;
  const int B = in_sizes[2];
  const int* src = ei;        // edge_index[0]
  const int* dst = ei + E;    // edge_index[1]

  // workspace carve-up (offsets stay 16B aligned for float4 access)
  char* ws = (char*)d_ws;
  int*   deg  = (int*)ws;    ws += (size_t)N * sizeof(int);
  float* dinv = (float*)ws;  ws += (size_t)N * sizeof(float);
  float* lin  = (float*)ws;  ws += (size_t)N * HIDDEN * sizeof(float);
  float* act  = (float*)ws;

  const int total  = N * HIDDEN;
  const int gN     = (N + 255) / 256;
  const int gE     = (E + 255) / 256;
  const int gTot   = (total + 255) / 256;
  const int gGemm  = (N / 16 + 7) / 8;
  const int gEdge  = (E + 7) / 8;
  const int gHead  = (B + 7) / 8;

  // normalization
  zero_deg_kernel<<<gN, 256, 0, stream>>>(deg, N);
  degree_kernel<<<gE, 256, 0, stream>>>(dst, E, deg);
  dinv_kernel<<<gN, 256, 0, stream>>>(deg, dinv, N);

  // layer 1
  gemm_wmma_kernel<<<gGemm, 256, 0, stream>>>(x, W1, lin, N);
  initagg_kernel<<<gTot, 256, 0, stream>>>(lin, dinv, act, total);
  edge_scatter_kernel<<<gEdge, 256, 0, stream>>>(src, dst, dinv, lin, act, E);
  finalize_kernel<<<gTot, 256, 0, stream>>>(act, b1, total);

  // layer 2 (reuse lin for the linear output, then overwrite act with agg2)
  gemm_wmma_kernel<<<gGemm, 256, 0, stream>>>(act, W2, lin, N);
  initagg_kernel<<<gTot, 256, 0, stream>>>(lin, dinv, act, total);
  edge_scatter_kernel<<<gEdge, 256, 0, stream>>>(src, dst, dinv, lin, act, E);
  finalize_kernel<<<gTot, 256, 0, stream>>>(act, b2, total);

  // head
  head_kernel<<<gHead, 256, 0, stream>>>(act, bidx, B, Wl, bl, out);
}